// LCGLayer_13357348290890
// MI455X (gfx1250) — compile-verified
//
#include <hip/hip_runtime.h>

// ---------------------------------------------------------------------------
// LCG layer: out[b,n,c] = sum_{e<32} sum_{i<4} x_pad[b, edges[n*32+e], i]
//                                            * weights[i, n*32+e, c]
// Per node: D(8x16) = A(8x128) x B(128x16) as 32 chained
// V_WMMA_F32_16X16X4_F32 ops (exact fp32, matches reference numerics).
//
// Memory-bound: 82 MB weight stream, read exactly once, 64B-coalesced.
// This version is branch-free in the hot loop and stages all 96 loads of a
// node into registers BEFORE the WMMA chain, so the wave keeps the HBM pipe
// full (staggered s_wait_loadcnt instead of drain-to-zero per chunk).
// ---------------------------------------------------------------------------

typedef __attribute__((ext_vector_type(2))) float v2f;
typedef __attribute__((ext_vector_type(8))) float v8f;

#define N_NODES  10000
#define DEG      32
#define CHANNELS 16
#define IN_DIM   4
#define BATCH    8
#define E_TOT    (N_NODES * DEG)
#define NODES_PER_BLOCK 4   // one wave32 per node

__global__ __launch_bounds__(NODES_PER_BLOCK * 32)
void lcg_wmma_f32_kernel(const float* __restrict__ x,      // (B, N, IN_DIM)
                         const int*   __restrict__ edges,  // (E,)
                         const float* __restrict__ w,      // (IN_DIM, E, C)
                         float*       __restrict__ out)    // (B, N, C)
{
    const int lane  = threadIdx.x & 31;
    const int wave  = threadIdx.x >> 5;
    const int node  = blockIdx.x * NODES_PER_BLOCK + wave;   // grid is exact

    const int m     = lane & 15;   // A row (batch b) / B col (channel c)
    const int khalf = lane >> 4;   // 0 -> K {0,1}, 1 -> K {2,3}
    const int i0    = khalf * 2;   // first in_dim index held by this lane

    // Lane mask for A rows 8..15 (unused batch rows) — applied by v_mul,
    // no EXEC divergence in the hot loop.
    const float laneMask = (m < BATCH) ? 1.0f : 0.0f;

    // Each lane owns one of the node's 32 edge indices; broadcast per chunk
    // with v_readlane (loop fully unrolled -> constant lane selects).
    const int myEdge = edges[node * DEG + lane];

    // B stream bases (hoisted 64-bit math; the i0+1 plane is 20.48 MB away,
    // beyond the 24-bit immediate, so keep it as a second base pointer).
    const float* bbase0 = w + ((size_t)i0 * E_TOT + (size_t)node * DEG) * CHANNELS + m;
    const float* bbase1 = bbase0 + (size_t)E_TOT * CHANNELS;

    // Per-lane constant part of the A gather address (batch row + in_dim).
    const int aLaneOff = (m & 7) * (N_NODES * IN_DIM) + i0;

    // ---- phase 1: issue ALL loads for this node, branch-free --------------
    v2f av[DEG];
    v2f bv[DEG];
#pragma unroll
    for (int e = 0; e < DEG; ++e) {
        const int  src   = __builtin_amdgcn_readlane(myEdge, e); // scalar
        const bool ok    = (src < N_NODES);                      // src==N -> pad(0)
        const int  src_c = ok ? src : 0;                         // scalar select
        const float mflag = ok ? laneMask : 0.0f;

        // A tile 16x4 f32: lane holds A[m][i0], A[m][i0+1] (8B load)
        v2f a = *(const v2f*)(x + (size_t)src_c * IN_DIM + aLaneOff);
        av[e].x = a.x * mflag;
        av[e].y = a.y * mflag;

        // B tile 4x16 f32: lane holds B[i0][c], B[i0+1][c]
        // (VGPR0: lanes0-15 row K=0, lanes16-31 row K=2; VGPR1: K=1 / K=3)
        bv[e].x = bbase0[e * CHANNELS];   // offset:e*64 immediate
        bv[e].y = bbase1[e * CHANNELS];
    }

    // ---- phase 2: 32 chained exact-fp32 WMMAs -----------------------------
    v8f acc = {0.f, 0.f, 0.f, 0.f, 0.f, 0.f, 0.f, 0.f};
#pragma unroll
    for (int e = 0; e < DEG; ++e) {
        acc = __builtin_amdgcn_wmma_f32_16x16x4_f32(
            /*neg_a=*/false, av[e], /*neg_b=*/false, bv[e],
            /*c_mod=*/(short)0, acc, /*reuse_a=*/false, /*reuse_b=*/false);
    }

    // ---- store: rows M=0..7 (batches) live in acc[0..7] on lanes 0-15 ----
    if (lane < 16) {
#pragma unroll
        for (int r = 0; r < BATCH; ++r) {
            out[(r * N_NODES + node) * CHANNELS + lane] = acc[r];
        }
    }
}

extern "C" void kernel_launch(void* const* d_in, const int* in_sizes, int n_in,
                              void* d_out, int out_size, void* d_ws, size_t ws_size,
                              hipStream_t stream) {
    (void)in_sizes; (void)n_in; (void)d_ws; (void)ws_size; (void)out_size;

    const float* x     = (const float*)d_in[0];  // (8, 10000, 4) f32
    const int*   edges = (const int*)  d_in[1];  // (320000,) int
    const float* w     = (const float*)d_in[2];  // (4, 320000, 16) f32
    float*       out   = (float*)d_out;          // (8, 10000, 16) f32

    const int blocks = N_NODES / NODES_PER_BLOCK;               // 2500 exact
    lcg_wmma_f32_kernel<<<blocks, NODES_PER_BLOCK * 32, 0, stream>>>(x, edges, w, out);
}